// GCN_87900800680759
// MI455X (gfx1250) — compile-verified
//
#include <hip/hip_runtime.h>

#define NN 10000
#define NE 160000
#define FIN 16
#define HEADS 8
#define CHN 8
#define HIDD 64
#define NGR 64
#define VOC 50000
#define LSTK 9
#define EP (NE + NN)   // edges + self loops

typedef __attribute__((ext_vector_type(2))) float v2f;
typedef __attribute__((ext_vector_type(8))) float v8f;

// ---- ordered-key mapping so float max can use unsigned atomicMax ----
__device__ __forceinline__ unsigned fkey(float f) {
  unsigned u = __float_as_uint(f);
  return (u >> 31) ? ~u : (u | 0x80000000u);
}
__device__ __forceinline__ float funkey(unsigned k) {
  return __uint_as_float((k >> 31) ? (k & 0x7FFFFFFFu) : ~k);
}
#define KEY_NEG_INF 0x007FFFFFu   // fkey(-inf)

// =====================================================================
// Fused triple GEMM: Ci[M x Nc] = A[M x K] * Bi[K x Nc] + bi  (i=0,1,2)
// One wave32 per 16x16 tile position; A fragments loaded once, three
// v_wmma_f32_16x16x4_f32 issued per k-step with independent accumulators.
// =====================================================================
__global__ void k_gemm3_wmma(const float* __restrict__ A,
                             const float* __restrict__ B0, const float* __restrict__ b0,
                             float* __restrict__ C0,
                             const float* __restrict__ B1, const float* __restrict__ b1,
                             float* __restrict__ C1,
                             const float* __restrict__ B2, const float* __restrict__ b2,
                             float* __restrict__ C2,
                             int K) {
  const int lane = threadIdx.x;        // 0..31
  const int half = lane >> 4;          // 0 or 1
  const int l    = lane & 15;
  const int tn = blockIdx.x, tm = blockIdx.y;
  const int arow = tm * 16 + l;
  const int bcol = tn * 16 + l;
  v8f acc0 = {}, acc1 = {}, acc2 = {};
  for (int k = 0; k < K; k += 4) {
    // A 16x4 fragment: lanes 0-15 hold K=k,k+1; lanes 16-31 hold K=k+2,k+3
    const float* ap = A + arow * K + k + half * 2;
    v2f a; a.x = ap[0]; a.y = ap[1];
    // B 4x16 fragments: lanes 0-15 rows k,k+1; lanes 16-31 rows k+2,k+3
    const int boff = (k + half * 2) * HIDD + bcol;
    v2f w0; w0.x = B0[boff]; w0.y = B0[boff + HIDD];
    v2f w1; w1.x = B1[boff]; w1.y = B1[boff + HIDD];
    v2f w2; w2.x = B2[boff]; w2.y = B2[boff + HIDD];
    acc0 = __builtin_amdgcn_wmma_f32_16x16x4_f32(false, a, false, w0, (short)0, acc0, false, false);
    acc1 = __builtin_amdgcn_wmma_f32_16x16x4_f32(false, a, false, w1, (short)0, acc1, false, false);
    acc2 = __builtin_amdgcn_wmma_f32_16x16x4_f32(false, a, false, w2, (short)0, acc2, false, false);
  }
  // C 16x16: VGPR r -> row (half*8 + r), col = lane%16
  const float bv0 = b0[bcol], bv1 = b1[bcol], bv2 = b2[bcol];
  const int rb = tm * 16 + half * 8;
#pragma unroll
  for (int r = 0; r < 8; ++r) {
    const int off = (rb + r) * HIDD + bcol;
    C0[off] = acc0[r] + bv0;
    C1[off] = acc1[r] + bv1;
    C2[off] = acc2[r] + bv2;
  }
}

// =====================================================================
// Single GEMM (classifier): C[M x Nc] = A[M x K] * B[K x Nc] + bias
// B rows are 200KB apart for the vocab GEMM -> prefetch next k-block.
// =====================================================================
__global__ void k_gemm_wmma(const float* __restrict__ A, const float* __restrict__ B,
                            const float* __restrict__ bias, float* __restrict__ C,
                            int M, int K, int Nc) {
  const int lane = threadIdx.x;
  const int half = lane >> 4;
  const int l    = lane & 15;
  const int tn = blockIdx.x, tm = blockIdx.y;
  const int arow = tm * 16 + l;
  const int bcol = tn * 16 + l;
  v8f acc = {};
  for (int k = 0; k < K; k += 4) {
    const float* ap = A + arow * K + k + half * 2;
    v2f a; a.x = ap[0]; a.y = ap[1];
    const float* bp = B + (size_t)(k + half * 2) * Nc + bcol;
    v2f b; b.x = bp[0]; b.y = bp[Nc];
    if (k + 4 < K) {                       // global_prefetch next k-block of B
      __builtin_prefetch(bp + 4 * (size_t)Nc, 0, 1);
      __builtin_prefetch(bp + 5 * (size_t)Nc, 0, 1);
    }
    acc = __builtin_amdgcn_wmma_f32_16x16x4_f32(false, a, false, b, (short)0, acc, false, false);
  }
  const float bv = bias[bcol];
  const int rb = tm * 16 + half * 8;
#pragma unroll
  for (int r = 0; r < 8; ++r)
    C[(size_t)(rb + r) * Nc + bcol] = acc[r] + bv;
}

// =====================================================================
// Graph prep: degree + mean incoming edge_attr (self-loop fill value)
// =====================================================================
__global__ void k_zero(float* __restrict__ p, int n) {
  int i = blockIdx.x * blockDim.x + threadIdx.x;
  if (i < n) p[i] = 0.f;
}

__global__ void k_deg_loop(const int* __restrict__ dstE, const float* __restrict__ ea,
                           float* __restrict__ deg, float* __restrict__ la) {
  int e = blockIdx.x * blockDim.x + threadIdx.x;
  if (e >= NE) return;
  int d = dstE[e];
  atomicAdd(deg + d, 1.f);
  atomicAdd(la + 2 * d,     ea[2 * e]);
  atomicAdd(la + 2 * d + 1, ea[2 * e + 1]);
}

__global__ void k_loop_div(float* __restrict__ la, const float* __restrict__ deg) {
  int i = blockIdx.x * blockDim.x + threadIdx.x;
  if (i < 2 * NN) la[i] /= fmaxf(deg[i >> 1], 1.f);
}

// =====================================================================
// Attention pipeline
// =====================================================================
__global__ void k_attn_init(unsigned* __restrict__ mxkey, float* __restrict__ den,
                            float* __restrict__ agg) {
  int i = blockIdx.x * blockDim.x + threadIdx.x;
  if (i < NN * HIDD) agg[i] = 0.f;
  if (i < NN * HEADS) { mxkey[i] = KEY_NEG_INF; den[i] = 0.f; }
}

__global__ void k_edge_logit(const float* __restrict__ xl, const float* __restrict__ xr,
                             const int* __restrict__ srcE, const int* __restrict__ dstE,
                             const float* __restrict__ ea, const float* __restrict__ la,
                             const float* __restrict__ We, const float* __restrict__ att,
                             float* __restrict__ logit, unsigned* __restrict__ mxkey) {
  int idx = blockIdx.x * blockDim.x + threadIdx.x;   // (edge, head)
  if (idx >= EP * HEADS) return;
  int e = idx >> 3, hh = idx & 7;
  int s, d; float a0, a1;
  if (e < NE) { s = srcE[e]; d = dstE[e]; a0 = ea[2 * e]; a1 = ea[2 * e + 1]; }
  else        { s = d = e - NE;           a0 = la[2 * s]; a1 = la[2 * s + 1]; }
  float lg = 0.f;
#pragma unroll
  for (int c = 0; c < CHN; ++c) {
    int hc = hh * CHN + c;
    float m = xl[s * HIDD + hc] + xr[d * HIDD + hc] + a0 * We[hc] + a1 * We[HIDD + hc];
    float g = m > 0.f ? m : 0.2f * m;                // leaky_relu(0.2)
    lg += g * att[hc];
  }
  logit[idx] = lg;
  atomicMax(mxkey + d * HEADS + hh, fkey(lg));       // segment max
}

__global__ void k_edge_exp(const float* __restrict__ logit, const unsigned* __restrict__ mxkey,
                           const int* __restrict__ dstE, float* __restrict__ exl,
                           float* __restrict__ den) {
  int idx = blockIdx.x * blockDim.x + threadIdx.x;
  if (idx >= EP * HEADS) return;
  int e = idx >> 3, hh = idx & 7;
  int d = (e < NE) ? dstE[e] : e - NE;
  float v = __expf(logit[idx] - funkey(mxkey[d * HEADS + hh]));
  exl[idx] = v;
  atomicAdd(den + d * HEADS + hh, v);                // segment sum
}

__global__ void k_edge_agg(const float* __restrict__ xl, const float* __restrict__ exl,
                           const float* __restrict__ den, const int* __restrict__ srcE,
                           const int* __restrict__ dstE, float* __restrict__ agg) {
  int idx = blockIdx.x * blockDim.x + threadIdx.x;
  if (idx >= EP * HEADS) return;
  int e = idx >> 3, hh = idx & 7;
  int s, d;
  if (e < NE) { s = srcE[e]; d = dstE[e]; } else { s = d = e - NE; }
  float alpha = exl[idx] / den[d * HEADS + hh];
#pragma unroll
  for (int c = 0; c < CHN; ++c)
    atomicAdd(agg + d * HIDD + hh * CHN + c, alpha * xl[s * HIDD + hh * CHN + c]);
}

__global__ void k_combine_elu(const float* __restrict__ agg, const float* __restrict__ bias_g,
                              const float* __restrict__ res, float* __restrict__ hout) {
  int idx = blockIdx.x * blockDim.x + threadIdx.x;
  if (idx >= NN * HIDD) return;
  float v = agg[idx] + bias_g[idx & (HIDD - 1)] + res[idx];
  hout[idx] = v > 0.f ? v : (__expf(v) - 1.f);       // elu
}

// =====================================================================
// Mean pooling per graph
// =====================================================================
__global__ void k_pool_accum(const float* __restrict__ h, const int* __restrict__ batch,
                             float* __restrict__ pooled, float* __restrict__ cnt) {
  int idx = blockIdx.x * blockDim.x + threadIdx.x;
  if (idx >= NN * HIDD) return;
  int n = idx >> 6, c = idx & (HIDD - 1);
  int b = batch[n];
  atomicAdd(pooled + b * HIDD + c, h[idx]);
  if (c == 0) atomicAdd(cnt + b, 1.f);
}

__global__ void k_pool_div(float* __restrict__ pooled, const float* __restrict__ cnt) {
  int idx = blockIdx.x * blockDim.x + threadIdx.x;
  if (idx < NGR * HIDD) pooled[idx] /= fmaxf(cnt[idx >> 6], 1.f);
}

// =====================================================================
extern "C" void kernel_launch(void* const* d_in, const int* in_sizes, int n_in,
                              void* d_out_v, int out_size, void* d_ws, size_t ws_size,
                              hipStream_t stream) {
  (void)in_sizes; (void)n_in; (void)out_size; (void)ws_size;
  const float* x       = (const float*)d_in[0];
  const int*   ei      = (const int*)  d_in[1];
  const float* ea      = (const float*)d_in[2];
  const int*   batch   = (const int*)  d_in[3];
  const float* p1_Wl   = (const float*)d_in[4];
  const float* p1_bl   = (const float*)d_in[5];
  const float* p1_Wr   = (const float*)d_in[6];
  const float* p1_br   = (const float*)d_in[7];
  const float* p1_We   = (const float*)d_in[8];
  const float* p1_att  = (const float*)d_in[9];
  const float* p1_bias = (const float*)d_in[10];
  const float* p1_Wres = (const float*)d_in[11];
  const float* p1_bres = (const float*)d_in[12];
  const float* Wl      = (const float*)d_in[13];
  const float* bl      = (const float*)d_in[14];
  const float* Wr      = (const float*)d_in[15];
  const float* br      = (const float*)d_in[16];
  const float* We      = (const float*)d_in[17];
  const float* att     = (const float*)d_in[18];
  const float* bias_g  = (const float*)d_in[19];
  const float* Wres    = (const float*)d_in[20];
  const float* bres    = (const float*)d_in[21];
  const float* Wlin    = (const float*)d_in[22];
  const float* blin    = (const float*)d_in[23];

  float* out = (float*)d_out_v;
  float* ws  = (float*)d_ws;

  // workspace layout (floats)
  float*    xl    = ws;                       // 640000
  float*    xr    = ws + 640000;              // 640000
  float*    res   = ws + 1280000;             // 640000
  float*    agg   = ws + 1920000;             // 640000
  float*    logit = ws + 2560000;             // EP*8 = 1360000
  float*    exl   = ws + 3920000;             // 1360000
  float*    den   = ws + 5280000;             // 80000
  unsigned* mxkey = (unsigned*)(ws + 5360000);// 80000
  float*    deg   = ws + 5440000;             // 10000
  float*    la    = ws + 5450000;             // 20000
  float*    cnt   = ws + 5470000;             // 64

  const int* srcE = ei;
  const int* dstE = ei + NE;

  const int TB = 256;
  const int gEdge = (EP * HEADS + TB - 1) / TB;
  const int gNode = (NN * HIDD + TB - 1) / TB;

  // ---- self-loop edge_attr (mean of incoming) ----
  k_zero<<<(NN + TB - 1) / TB, TB, 0, stream>>>(deg, NN);
  k_zero<<<(2 * NN + TB - 1) / TB, TB, 0, stream>>>(la, 2 * NN);
  k_deg_loop<<<(NE + TB - 1) / TB, TB, 0, stream>>>(dstE, ea, deg, la);
  k_loop_div<<<(2 * NN + TB - 1) / TB, TB, 0, stream>>>(la, deg);

  // ---- 10 GATv2 layers; layer i output lands in d_out slice i ----
  for (int layer = 0; layer < 1 + LSTK; ++layer) {
    const float* hin = (layer == 0) ? x : out + (size_t)(layer - 1) * NN * HIDD;
    const int K = (layer == 0) ? FIN : HIDD;
    const int li = layer - 1;
    const float* Wl_   = (layer == 0) ? p1_Wl   : Wl   + (size_t)li * HIDD * HIDD;
    const float* bl_   = (layer == 0) ? p1_bl   : bl   + (size_t)li * HIDD;
    const float* Wr_   = (layer == 0) ? p1_Wr   : Wr   + (size_t)li * HIDD * HIDD;
    const float* br_   = (layer == 0) ? p1_br   : br   + (size_t)li * HIDD;
    const float* We_   = (layer == 0) ? p1_We   : We   + (size_t)li * 2 * HIDD;
    const float* att_  = (layer == 0) ? p1_att  : att  + (size_t)li * HEADS * CHN;
    const float* bg_   = (layer == 0) ? p1_bias : bias_g + (size_t)li * HIDD;
    const float* Wres_ = (layer == 0) ? p1_Wres : Wres + (size_t)li * HIDD * HIDD;
    const float* bres_ = (layer == 0) ? p1_bres : bres + (size_t)li * HIDD;
    float* hout = out + (size_t)layer * NN * HIDD;

    // fused xl/xr/res transforms: A read once, 3 WMMA streams
    dim3 gg(HIDD / 16, NN / 16);   // (4, 625) 16x16 tiles
    k_gemm3_wmma<<<gg, 32, 0, stream>>>(hin, Wl_, bl_, xl, Wr_, br_, xr,
                                        Wres_, bres_, res, K);

    k_attn_init<<<gNode, TB, 0, stream>>>(mxkey, den, agg);
    k_edge_logit<<<gEdge, TB, 0, stream>>>(xl, xr, srcE, dstE, ea, la, We_, att_, logit, mxkey);
    k_edge_exp  <<<gEdge, TB, 0, stream>>>(logit, mxkey, dstE, exl, den);
    k_edge_agg  <<<gEdge, TB, 0, stream>>>(xl, exl, den, srcE, dstE, agg);
    k_combine_elu<<<gNode, TB, 0, stream>>>(agg, bg_, res, hout);
  }

  // ---- mean pool + classifier ----
  float* pooled = out + (size_t)(1 + LSTK) * NN * HIDD;   // 6400000
  float* x12    = pooled + NGR * HIDD;                    // 6404096
  k_zero<<<(NGR * HIDD + TB - 1) / TB, TB, 0, stream>>>(pooled, NGR * HIDD);
  k_zero<<<1, NGR, 0, stream>>>(cnt, NGR);
  k_pool_accum<<<gNode, TB, 0, stream>>>(out + (size_t)LSTK * NN * HIDD, batch, pooled, cnt);
  k_pool_div<<<(NGR * HIDD + TB - 1) / TB, TB, 0, stream>>>(pooled, cnt);

  dim3 gf(VOC / 16, NGR / 16);     // (3125, 4)
  k_gemm_wmma<<<gf, 32, 0, stream>>>(pooled, Wlin, blin, x12, NGR, HIDD, VOC);
}